// MoE_12317966205421
// MI455X (gfx1250) — compile-verified
//
#include <hip/hip_runtime.h>

typedef __attribute__((ext_vector_type(2))) float v2f;
typedef __attribute__((ext_vector_type(8))) float v8f;

#define E_   8
#define TOPK 2
#define C_   128
#define H_   96
#define W_   96
#define B_   8
#define NG   4
#define HW_  (H_*W_)
#define OUT1 (B_*C_*HW_)   // one gate's output tensor
#define NWC  (E_*9*C_*C_)  // packed conv weights
#define NWP  (E_*C_*C_)    // packed pointwise weights

#define PATCH (C_*3*34)    // 13056 floats: [ci][dh 0..2][wi 0..33]

// ---------------- zero output ----------------
__global__ void moe_zero_kernel(float* __restrict__ p, long n) {
  long i = (long)blockIdx.x * blockDim.x + threadIdx.x;
  long stride = (long)gridDim.x * blockDim.x;
  for (; i < n; i += stride) p[i] = 0.f;
}

// ------------- weight packing -------------
// wct[(((e*9 + r)*32 + cb)*C + co)*4 + q] = Wc[(((e*C+co)*C + cb*4+q)*9) + r]
//   -> lane K-pair (ci = c0+kl, c0+kl+1) is 8 contiguous bytes => global_load_b64
// wpt[((e*32 + cb)*C + o)*4 + q]          = Wp[(e*C+o)*C + cb*4+q]
__global__ void moe_pack_kernel(const float* __restrict__ Wc,
                                const float* __restrict__ Wp,
                                float* __restrict__ wct,
                                float* __restrict__ wpt) {
  int i = blockIdx.x * blockDim.x + threadIdx.x;
  int stride = gridDim.x * blockDim.x;
  for (int idx = i; idx < NWC + NWP; idx += stride) {
    if (idx < NWC) {
      int t = idx;
      int q  = t & 3;   t >>= 2;
      int co = t & 127; t >>= 7;
      int cb = t & 31;  t >>= 5;
      int r  = t % 9;   int e = t / 9;
      wct[idx] = Wc[(((e*C_ + co)*C_ + (cb*4 + q))*9) + r];
    } else {
      int t = idx - NWC;
      int q  = t & 3;   int t2 = t >> 2;
      int o  = t2 & 127; t2 >>= 7;
      int cb = t2 & 31;  int e = t2 >> 5;
      wpt[t] = Wp[(e*C_ + o)*C_ + (cb*4 + q)];
    }
  }
}

// ------------- global average pool -------------
__global__ void moe_gap_kernel(const float* __restrict__ x, float* __restrict__ gap) {
  __shared__ float red[256];
  int bc = blockIdx.x;                    // b*C + c
  const float* p = x + (long)bc * HW_;
  float s = 0.f;
  for (int i = threadIdx.x; i < HW_; i += 256) s += p[i];
  red[threadIdx.x] = s;
  __syncthreads();
  for (int off = 128; off > 0; off >>= 1) {
    if ((int)threadIdx.x < off) red[threadIdx.x] += red[threadIdx.x + off];
    __syncthreads();
  }
  if (threadIdx.x == 0) gap[bc] = red[0] * (1.f / (float)HW_);
}

// ------------- gating: softmax, top2, cv^2 loss -------------
__global__ void moe_gate_kernel(const float* __restrict__ G,
                                const float* __restrict__ gap,
                                float* __restrict__ tw, int* __restrict__ ti,
                                float* __restrict__ loss_out) {
  __shared__ float probs[NG][B_][E_];
  __shared__ float imp[NG][E_];
  int t = threadIdx.x;                   // 256 = 4*8*8
  {
    int e = t & 7, b = (t >> 3) & 7, g = t >> 6;
    float s = 0.f;
    for (int c = 0; c < C_; ++c) s += gap[b*C_ + c] * G[(g*C_ + c)*E_ + e];
    probs[g][b][e] = s;
  }
  __syncthreads();
  if (t < NG * B_) {
    int b = t & 7, g = t >> 3;
    float m = -1e30f;
    for (int e = 0; e < E_; ++e) m = fmaxf(m, probs[g][b][e]);
    float pe[E_]; float denom = 0.f;
    for (int e = 0; e < E_; ++e) { pe[e] = expf(probs[g][b][e] - m); denom += pe[e]; }
    float inv = 1.f / denom;
    for (int e = 0; e < E_; ++e) { pe[e] *= inv; probs[g][b][e] = pe[e]; }
    // top-2, jax tie-breaking (lowest index wins ties)
    int i0 = 0;
    for (int e = 1; e < E_; ++e) if (pe[e] > pe[i0]) i0 = e;
    int i1 = (i0 == 0) ? 1 : 0;
    for (int e = 0; e < E_; ++e) { if (e == i0) continue; if (pe[e] > pe[i1]) i1 = e; }
    float r = expf(pe[i1] - pe[i0]);       // softmax over the two top probabilities
    float w0 = 1.f / (1.f + r), w1 = r / (1.f + r);
    ti[(g*B_ + b)*2 + 0] = i0;  ti[(g*B_ + b)*2 + 1] = i1;
    tw[(g*B_ + b)*2 + 0] = w0;  tw[(g*B_ + b)*2 + 1] = w1;
  }
  __syncthreads();
  if (t < NG * E_) {
    int e = t & 7, g = t >> 3;
    float s = 0.f;
    for (int b = 0; b < B_; ++b) s += probs[g][b][e];
    imp[g][e] = s;
  }
  __syncthreads();
  if (t == 0) {
    float loss = 0.f;
    for (int g = 0; g < NG; ++g) {
      float mean = 0.f;
      for (int e = 0; e < E_; ++e) mean += imp[g][e];
      mean *= (1.f / (float)E_);
      float var = 0.f;
      for (int e = 0; e < E_; ++e) { float d = imp[g][e] - mean; var += d * d; }
      var *= (1.f / (float)(E_ - 1));     // ddof=1
      loss += var / (mean * mean + 1e-10f);
    }
    *loss_out = loss * (1.f / (float)NG);
  }
}

// ------------- fused expert: conv3x3 + relu + squash + pointwise + combine -------------
// Workgroup: 8 waves, tile = 128 co x 32 pixels (one row, w0..w0+31).
// Each wave: one 16-co stripe, two 16-pixel N tiles sharing every A fragment.
__global__ __launch_bounds__(256, 1)
void moe_expert_kernel(const float* __restrict__ x,  const float* __restrict__ wct,
                       const float* __restrict__ wpt, const float* __restrict__ bc,
                       const float* __restrict__ bp,  const float* __restrict__ tw,
                       const int* __restrict__ ti,    float* __restrict__ out) {
  const int e  = blockIdx.z;
  const int b  = blockIdx.y;
  const int pb = blockIdx.x;              // 0..287: h = pb/3, w0 = (pb%3)*32

  // combined gate weight per gate for this (e,b); skip block if never selected
  float wsum[NG];
  bool any = false;
#pragma unroll
  for (int g = 0; g < NG; ++g) {
    float w = 0.f;
#pragma unroll
    for (int k = 0; k < TOPK; ++k)
      if (ti[(g*B_ + b)*2 + k] == e) w += tw[(g*B_ + b)*2 + k];
    wsum[g] = w;
    any = any || (w != 0.f);
  }
  if (!any) return;

  const int h  = pb / 3;
  const int w0 = (pb % 3) * 32;

  __shared__ float smem[PATCH];           // patch, later aliased as us[C][32]
  __shared__ float sq[32];                // per-pixel sum of u^2

  // stage zero-padded x patch: [ci][dh][wi], wi in 0..33  (51 elems/thread)
  for (int idx = threadIdx.x; idx < PATCH; idx += 256) {
    int ci = idx / 102, rem = idx % 102;
    int dh = rem / 34,  wi = rem % 34;
    int hin = h + dh - 1, win = w0 + wi - 1;
    float v = 0.f;
    if (hin >= 0 && hin < H_ && win >= 0 && win < W_)
      v = x[((long)(b*C_ + ci)*H_ + hin)*W_ + win];
    smem[idx] = v;
  }
  if (threadIdx.x < 32) sq[threadIdx.x] = 0.f;
  __syncthreads();

  const int lane = threadIdx.x & 31;
  const int wave = threadIdx.x >> 5;
  const int ml   = lane & 15;             // M lane for A, N (pixel) lane for B/D
  const int hi   = lane >> 4;             // upper half: K pair 2..3 / rows M+8
  const int kl   = hi * 2;
  const int co0  = wave * 16;

  // ---- conv: 9 shifted GEMMs, D = two 16x16 tiles, K = 9*128 ----
  v8f acc0 = {0.f,0.f,0.f,0.f,0.f,0.f,0.f,0.f};
  v8f acc1 = {0.f,0.f,0.f,0.f,0.f,0.f,0.f,0.f};
  const float* wce = wct + ((long)e * 9) * (32 * C_ * 4);
#pragma unroll
  for (int kh = 0; kh < 3; ++kh) {
#pragma unroll
    for (int kw = 0; kw < 3; ++kw) {
      // per-lane A base: packed [cb][co][q], lane K-pair contiguous (b64 loads)
      const float* wA = wce + (kh*3 + kw) * (32 * C_ * 4) + (co0 + ml) * 4 + kl;
      const float* xB = smem + kh*34 + (ml + kw);
      for (int cb = 0; cb < 32; ++cb) {
        v2f a = *(const v2f*)(wA + cb * (C_ * 4));       // global_load_b64
        int ci = cb * 4 + kl;
        v2f b0, b1;
        b0.x = xB[ ci      * 102];
        b0.y = xB[(ci + 1) * 102];
        b1.x = xB[ ci      * 102 + 16];
        b1.y = xB[(ci + 1) * 102 + 16];
        acc0 = __builtin_amdgcn_wmma_f32_16x16x4_f32(
                 false, a, false, b0, (short)0, acc0, false, false);
        acc1 = __builtin_amdgcn_wmma_f32_16x16x4_f32(
                 false, a, false, b1, (short)0, acc1, false, false);
      }
    }
  }

  // bias + relu + per-pixel sum of squares (cross-wave via LDS atomics)
  float p0s = 0.f, p1s = 0.f;
#pragma unroll
  for (int r = 0; r < 8; ++r) {
    float bias = bc[e*C_ + co0 + r + hi*8];
    float v0 = fmaxf(acc0[r] + bias, 0.f);
    float v1 = fmaxf(acc1[r] + bias, 0.f);
    acc0[r] = v0; acc1[r] = v1;
    p0s += v0 * v0;
    p1s += v1 * v1;
  }
  atomicAdd(&sq[ml],      p0s);
  atomicAdd(&sq[ml + 16], p1s);
  __syncthreads();                        // also fences patch reads before reuse

  const float sn0 = sq[ml], sn1 = sq[ml + 16];
  const float sc0 = sn0 / (1.f + sn0) / sqrtf(sn0 + 1e-8f);
  const float sc1 = sn1 / (1.f + sn1) / sqrtf(sn1 + 1e-8f);
  float* us = smem;                       // alias: [c][pix 0..31]
#pragma unroll
  for (int r = 0; r < 8; ++r) {
    int c = co0 + r + hi*8;
    us[c*32 + ml]      = acc0[r] * sc0;
    us[c*32 + ml + 16] = acc1[r] * sc1;
  }
  __syncthreads();

  // ---- pointwise: out[o,pix] = sum_c Wp[o,c] * u[c,pix], K = 128 ----
  v8f d0 = {0.f,0.f,0.f,0.f,0.f,0.f,0.f,0.f};
  v8f d1 = {0.f,0.f,0.f,0.f,0.f,0.f,0.f,0.f};
  const float* wpe = wpt + (long)e * (32 * C_ * 4) + (co0 + ml) * 4 + kl;
  for (int cb = 0; cb < 32; ++cb) {
    v2f a = *(const v2f*)(wpe + cb * (C_ * 4));          // global_load_b64
    int ci = cb * 4 + kl;
    v2f b0, b1;
    b0.x = us[ ci      * 32 + ml];
    b0.y = us[(ci + 1) * 32 + ml];
    b1.x = us[ ci      * 32 + ml + 16];
    b1.y = us[(ci + 1) * 32 + ml + 16];
    d0 = __builtin_amdgcn_wmma_f32_16x16x4_f32(
           false, a, false, b0, (short)0, d0, false, false);
    d1 = __builtin_amdgcn_wmma_f32_16x16x4_f32(
           false, a, false, b1, (short)0, d1, false, false);
  }

  // bias + weighted scatter into selected gates' outputs
  const int p0 = h * W_ + w0;
#pragma unroll
  for (int r = 0; r < 8; ++r) {
    int c = co0 + r + hi*8;
    float bias = bp[e*C_ + c];
    float v0 = d0[r] + bias;
    float v1 = d1[r] + bias;
    long base = (long)(b*C_ + c)*HW_ + p0 + ml;
#pragma unroll
    for (int g = 0; g < NG; ++g)
      if (wsum[g] != 0.f) {
        atomicAdd(out + (long)g*OUT1 + base,      wsum[g] * v0);
        atomicAdd(out + (long)g*OUT1 + base + 16, wsum[g] * v1);
      }
  }
}

extern "C" void kernel_launch(void* const* d_in, const int* in_sizes, int n_in,
                              void* d_out, int out_size, void* d_ws, size_t ws_size,
                              hipStream_t stream) {
  const float* x  = (const float*)d_in[0];
  const float* G  = (const float*)d_in[1];
  const float* Wc = (const float*)d_in[2];
  const float* bc = (const float*)d_in[3];
  const float* Wp = (const float*)d_in[4];
  const float* bp = (const float*)d_in[5];
  float* out = (float*)d_out;

  float* ws  = (float*)d_ws;
  float* wct = ws;                 // NWC floats
  float* wpt = wct + NWC;          // NWP floats
  float* gap = wpt + NWP;          // B*C
  float* tw  = gap + B_*C_;        // NG*B*2
  int*   ti  = (int*)(tw + NG*B_*2);

  const long nout = (long)NG * OUT1 + 1;
  moe_zero_kernel<<<2048, 256, 0, stream>>>(out, nout);
  moe_pack_kernel<<<2048, 256, 0, stream>>>(Wc, Wp, wct, wpt);
  moe_gap_kernel<<<B_*C_, 256, 0, stream>>>(x, gap);
  moe_gate_kernel<<<1, 256, 0, stream>>>(G, gap, tw, ti, out + (long)NG*OUT1);

  dim3 grid(H_ * 3, B_, E_);       // 288 pixel tiles (32 wide) x 8 batch x 8 experts
  moe_expert_kernel<<<grid, 256, 0, stream>>>(x, wct, wpt, bc, bp, tw, ti, out);
}